// BPSEncoder_62414464745827
// MI455X (gfx1250) — compile-verified
//
#include <hip/hip_runtime.h>
#include <hip/hip_bf16.h>

typedef float v2f __attribute__((ext_vector_type(2)));
typedef float v8f __attribute__((ext_vector_type(8)));
typedef unsigned int v4u __attribute__((ext_vector_type(4)));
typedef int v4i __attribute__((ext_vector_type(4)));
typedef int v8i __attribute__((ext_vector_type(8)));

#define LEAKY_SLOPE 0.2f
#define GEMM_CHUNK 64   // W rows staged per TDM transfer (must divide slab_k)

// ---------------------------------------------------------------------------
// Kernel 1: fused BPS encode (nearest neighbor + dist/delta write).
// Grid: (K/256, B), block 256. Each thread owns one basis point k, streams the
// whole point cloud of batch b through LDS (float4 per point -> single
// ds_load_b128 broadcast per candidate), tracks (min d2, argmin n).
// ---------------------------------------------------------------------------
__global__ __launch_bounds__(256) void bps_encode_kernel(
    const float* __restrict__ pc,      // [B, N, 3]
    const float* __restrict__ basis,   // [K, 3]
    float*       __restrict__ bps,     // [B, 4K]  = [dists K | deltas 3K]
    int N, int K)
{
    __shared__ float4 spts[256];
    const int t = threadIdx.x;
    const int k = blockIdx.x * 256 + t;
    const int b = blockIdx.y;

    const float bx = basis[k * 3 + 0];
    const float by = basis[k * 3 + 1];
    const float bz = basis[k * 3 + 2];

    const float* __restrict__ pcb = pc + (size_t)b * N * 3;

    float best = 3.4e38f;
    int   bestn = 0;

    for (int n0 = 0; n0 < N; n0 += 256) {
        __syncthreads();
        const int n = n0 + t;
        float px = pcb[n * 3 + 0];
        float py = pcb[n * 3 + 1];
        float pz = pcb[n * 3 + 2];
        spts[t] = make_float4(px, py, pz, 0.0f);
        __syncthreads();
#pragma unroll 8
        for (int j = 0; j < 256; ++j) {
            float4 p = spts[j];
            float dx = p.x - bx;
            float dy = p.y - by;
            float dz = p.z - bz;
            float d2 = fmaf(dx, dx, fmaf(dy, dy, dz * dz));
            bool c = d2 < best;              // strict <: first index wins (argmin)
            best  = c ? d2 : best;
            bestn = c ? (n0 + j) : bestn;
        }
    }

    const float nx = pcb[bestn * 3 + 0];
    const float ny = pcb[bestn * 3 + 1];
    const float nz = pcb[bestn * 3 + 2];

    float* __restrict__ row = bps + (size_t)b * (4 * K);
    row[k]                 = sqrtf(best);
    row[K + 3 * k + 0]     = nx - bx;
    row[K + 3 * k + 1]     = ny - by;
    row[K + 3 * k + 2]     = nz - bz;
}

// ---------------------------------------------------------------------------
// TDM descriptor build + issue: load a [rows x 16] fp32 tile of W (row-major,
// row stride N dwords) starting at row kc / col tile*16 into packed LDS.
// D# layout per CDNA5 ISA 8.3/8.4 (group0: flags/lds/global addr, group1:
// data_size=4B, tensor dims, tile dims, dim0 stride).
// This toolchain exposes the 6-arg builtin:
//   (uint32x4 g0, int32x8 g1, int32x4 g2, int32x4 g3, int32x8 gx, i32 cpol)
// ---------------------------------------------------------------------------
#if __has_builtin(__builtin_amdgcn_tensor_load_to_lds)
#define USE_TDM 1
__device__ __forceinline__ void tdm_load_wtile(
    const float* W, int K, int N, int kc, int col0, unsigned int lds_byte_off,
    int rows)
{
    unsigned long long gaddr =
        (unsigned long long)(uintptr_t)(W + (size_t)kc * N + col0);
    v4u g0;
    g0.x = 1u;                                   // count=1, user descriptor
    g0.y = lds_byte_off;                         // LDS byte address
    g0.z = (unsigned int)gaddr;                  // global_addr[31:0]
    g0.w = (unsigned int)(gaddr >> 32) | (2u << 30);  // addr[56:32] | type=2
    v8i g1;
    g1[0] = (2 << 16);                           // data_size = 4 bytes
    g1[1] = (int)((N & 0xFFFF) << 16);           // tensor_dim0[15:0]
    g1[2] = (int)(((unsigned)N >> 16) & 0xFFFF)  // tensor_dim0[31:16]
          | (int)((K & 0xFFFF) << 16);           // tensor_dim1[15:0]
    g1[3] = (int)(((unsigned)K >> 16) & 0xFFFF)  // tensor_dim1[31:16]
          | (16 << 16);                          // tile_dim0 = 16 cols
    g1[4] = rows;                                // tile_dim1 = rows, tile_dim2=0
    g1[5] = N;                                   // tensor_dim0_stride[31:0]
    g1[6] = 0;                                   // stride0 hi | dim1_stride lo
    g1[7] = 0;
    v4i z4 = {0, 0, 0, 0};
    v8i z8 = {0, 0, 0, 0, 0, 0, 0, 0};
    __builtin_amdgcn_tensor_load_to_lds(g0, g1, z4, z4, z8, 0);
}
#else
#define USE_TDM 0
#endif

// ---------------------------------------------------------------------------
// Kernel 2: split-K WMMA GEMM partial.  C_tile += A(16 x slab_k) * W(slab_k x 16)
// using V_WMMA_F32_16X16X4_F32.  A is [16,K] row-major, W is [K,N] row-major.
// Grid: (N/16 tiles, slabs), block = 32 (one wave, EXEC all ones for WMMA).
// W tiles are DMA'd into LDS via the Tensor Data Mover, double-buffered with
// s_wait_tensorcnt (TENSORcnt is in-order per wave, so waiting <=1 after
// issuing chunk c+1 guarantees chunk c is resident).
//
// Fragment layouts (ISA 7.12.2):
//   A 16x4 f32 : lane l -> M = l&15 ; VGPR0 holds K = (l<16 ? 0 : 2),
//                VGPR1 holds K = (l<16 ? 1 : 3)
//   B 4x16 f32 : VGPR v -> lanes 0-15: row K=v, lanes 16-31: row K=v+2
//   C 16x16    : VGPR v -> lanes 0-15: (M=v, N=lane), lanes 16-31: (M=v+8)
// ---------------------------------------------------------------------------
__global__ __launch_bounds__(32) void gemm_wmma_partial_kernel(
    const float* __restrict__ A,       // [16, K] row-major
    const float* __restrict__ W,       // [K, N] row-major
    float*       __restrict__ partial, // [slabs][Ntiles][256]
    int K, int N, int slab_k)
{
    const int tile = blockIdx.x;
    const int slab = blockIdx.y;
    const int lane = threadIdx.x;

    const int m     = lane & 15;
    const int khalf = (lane >> 4) << 1;          // 0 for lanes 0-15, 2 for 16-31
    const int colin = lane & 15;
    const int col0  = tile << 4;

    const int k0 = slab * slab_k;

    const float* __restrict__ Arow = A + (size_t)m * K + khalf;

    v8f acc = {};

#if USE_TDM
    __shared__ float wbuf[2][GEMM_CHUNK * 16];   // 8 KB double buffer
    const int nchunks = slab_k / GEMM_CHUNK;

    tdm_load_wtile(W, K, N, k0, col0,
                   (unsigned int)(uintptr_t)&wbuf[0][0], GEMM_CHUNK);

    for (int c = 0; c < nchunks; ++c) {
        if (c + 1 < nchunks) {
            tdm_load_wtile(W, K, N, k0 + (c + 1) * GEMM_CHUNK, col0,
                           (unsigned int)(uintptr_t)&wbuf[(c + 1) & 1][0],
                           GEMM_CHUNK);
            __builtin_amdgcn_s_wait_tensorcnt(1);   // chunk c resident
        } else {
            __builtin_amdgcn_s_wait_tensorcnt(0);   // last chunk resident
        }
        const float* __restrict__ wb = &wbuf[c & 1][0];
        const int kb = k0 + c * GEMM_CHUNK;
#pragma unroll 4
        for (int kk = 0; kk < GEMM_CHUNK; kk += 4) {
            v2f a = *(const v2f*)(Arow + kb + kk);          // global_load_b64
            v2f bf;                                          // ds_load_b32 x2
            bf.x = wb[(kk + khalf) * 16 + colin];
            bf.y = wb[(kk + khalf + 1) * 16 + colin];
            acc = __builtin_amdgcn_wmma_f32_16x16x4_f32(
                false, a, false, bf, (short)0, acc, false, false);
        }
    }
#else
    // Fallback: stream W directly from global.
    const float* __restrict__ Wcol = W + (size_t)khalf * N + col0 + colin;
    const int k1 = k0 + slab_k;
    for (int k = k0; k < k1; k += 4) {
        v2f a = *(const v2f*)(Arow + k);
        v2f bf;
        bf.x = Wcol[(size_t)k * N];
        bf.y = Wcol[(size_t)(k + 1) * N];
        if (k + 64 < K) __builtin_prefetch(&Wcol[(size_t)(k + 64) * N], 0, 1);
        acc = __builtin_amdgcn_wmma_f32_16x16x4_f32(
            false, a, false, bf, (short)0, acc, false, false);
    }
#endif

    float* out = partial + ((size_t)slab * gridDim.x + tile) * 256 + lane * 8;
    *(v8f*)out = acc;
}

// ---------------------------------------------------------------------------
// Kernel 3: reduce split-K partials, add bias, optional leaky-ReLU.
// One thread per output element of the [16, N] activation.
// ---------------------------------------------------------------------------
__global__ __launch_bounds__(256) void reduce_bias_act_kernel(
    const float* __restrict__ partial, // [slabs][Ntiles][256]
    const float* __restrict__ bias,    // [N]
    float*       __restrict__ out,     // [16, N] row-major
    int N, int slabs, int do_leaky)
{
    const int idx = blockIdx.x * 256 + threadIdx.x;
    if (idx >= 16 * N) return;
    const int r = idx / N;
    const int c = idx % N;

    const int tile   = c >> 4;
    const int lane   = (c & 15) + ((r >= 8) ? 16 : 0);
    const int v      = r & 7;
    const int Ntiles = N >> 4;

    float s = bias[c];
    for (int sl = 0; sl < slabs; ++sl) {
        s += partial[((size_t)sl * Ntiles + tile) * 256 + lane * 8 + v];
    }
    if (do_leaky) s = (s >= 0.0f) ? s : LEAKY_SLOPE * s;
    out[idx] = s;
}

// ---------------------------------------------------------------------------
extern "C" void kernel_launch(void* const* d_in, const int* in_sizes, int n_in,
                              void* d_out, int out_size, void* d_ws, size_t ws_size,
                              hipStream_t stream) {
    const float* pc    = (const float*)d_in[0];  // [16,4096,3]
    const float* basis = (const float*)d_in[1];  // [4096,3]
    const float* W1    = (const float*)d_in[2];  // [16384,1024]
    const float* b1    = (const float*)d_in[3];  // [1024]
    const float* W2    = (const float*)d_in[4];  // [1024,512]
    const float* b2    = (const float*)d_in[5];  // [512]
    const float* W3    = (const float*)d_in[6];  // [512,256]
    const float* b3    = (const float*)d_in[7];  // [256]

    const int B = 16, N = 4096, K = 4096;
    const int IN = 4 * K;        // 16384
    const int H1 = 1024, H2 = 512, OUT = 256;

    float* out = (float*)d_out;
    float* gf  = out;                 // [16, 256]
    float* bps = out + B * OUT;       // [16, 16384] (second tuple output)

    float* ws   = (float*)d_ws;
    float* h1   = ws;                         // 16*1024
    float* h2   = ws + B * H1;                // 16*512
    float* part = ws + B * H1 + B * H2;       // up to 16*64*256 floats (1 MB)

    // 1) BPS encode: 256 blocks x 256 threads, fused NN search.
    bps_encode_kernel<<<dim3(K / 256, B), 256, 0, stream>>>(pc, basis, bps, N, K);

    // 2) MLP layer 1: [16,16384] x [16384,1024]; 64 tiles x 16 K-slabs of 1024.
    gemm_wmma_partial_kernel<<<dim3(H1 / 16, 16), 32, 0, stream>>>(
        bps, W1, part, IN, H1, IN / 16);
    reduce_bias_act_kernel<<<dim3((B * H1 + 255) / 256), 256, 0, stream>>>(
        part, b1, h1, H1, 16, 1);

    // 3) MLP layer 2: [16,1024] x [1024,512]; 32 tiles x 8 K-slabs of 128.
    gemm_wmma_partial_kernel<<<dim3(H2 / 16, 8), 32, 0, stream>>>(
        h1, W2, part, H1, H2, H1 / 8);
    reduce_bias_act_kernel<<<dim3((B * H2 + 255) / 256), 256, 0, stream>>>(
        part, b2, h2, H2, 8, 1);

    // 4) MLP layer 3: [16,512] x [512,256]; 16 tiles x 4 K-slabs of 128.
    gemm_wmma_partial_kernel<<<dim3(OUT / 16, 4), 32, 0, stream>>>(
        h2, W3, part, H2, OUT, H2 / 4);
    reduce_bias_act_kernel<<<dim3((B * OUT + 255) / 256), 256, 0, stream>>>(
        part, b3, gf, OUT, 4, 0);
}